// GCN_82652350644770
// MI455X (gfx1250) — compile-verified
//
#include <hip/hip_runtime.h>
#include <math.h>

typedef float v2f __attribute__((ext_vector_type(2)));
typedef float v8f __attribute__((ext_vector_type(8)));

#define DIM 128

// ---------------------------------------------------------------------------
// Kernel 1: fused dual GEMM  h = seq@W_fc ;  out = seq@W_proj + bias + b_proj
// Block = 256 threads = 8 waves; each wave computes a 16-row stripe.
// WMMA f32 16x16x4: A (16x4) layout -> lane m=lane&15, lanes0-15 hold K=0,1
// in the two A VGPRs, lanes16-31 hold K=2,3. B (4x16): lane n=lane&15,
// lanes0-15 hold K-rows 0,1, lanes16-31 K-rows 2,3 (mirrors C/D half-split).
// C/D (16x16): VGPR v -> row v (lanes0-15) / row v+8 (lanes16-31), col=lane&15.
// ---------------------------------------------------------------------------
__global__ __launch_bounds__(256) void gcn_gemm_wmma(
    const float* __restrict__ seq,
    const float* __restrict__ w_fc,
    const float* __restrict__ w_proj,
    const float* __restrict__ bias,
    const float* __restrict__ b_proj,
    float* __restrict__ h,
    float* __restrict__ out,
    int n)
{
    const int lane = threadIdx.x & 31;
    const int wave = threadIdx.x >> 5;
    const int row0 = blockIdx.x * 128 + wave * 16;

    const int m  = lane & 15;          // A row within tile
    const int kh = (lane >> 4) * 2;    // K sub-offset (0 or 2) within 4-chunk
    const int nn = lane & 15;          // B/C/D column within tile

    // Hoist this wave's 16x128 A tile into registers: 32 x float2 per lane.
    int arow = row0 + m;
    if (arow >= n) arow = n - 1;       // clamp; bogus rows masked at store
    const float* abase = seq + (size_t)arow * DIM + kh;
    v2f a[32];
#pragma unroll
    for (int kk = 0; kk < 32; ++kk)
        a[kk] = *(const v2f*)(abase + kk * 4);

    const int rowsel = (lane >> 4) << 3;   // +8 for upper half-lanes at store

#pragma unroll 1
    for (int t = 0; t < 8; ++t) {
        const int colBase = t * 16;
        v8f cf = {};   // seq @ W_fc accumulator
        v8f cp = {};   // seq @ W_proj accumulator

        const float* bf = w_fc   + (size_t)kh * DIM + colBase + nn;
        const float* bp = w_proj + (size_t)kh * DIM + colBase + nn;

#pragma unroll
        for (int kk = 0; kk < 32; ++kk) {
            const size_t r0 = (size_t)(kk * 4) * DIM;
            const size_t r1 = (size_t)(kk * 4 + 1) * DIM;
            v2f bfr, bpr;
            bfr.x = bf[r0];  bfr.y = bf[r1];
            bpr.x = bp[r0];  bpr.y = bp[r1];
            cf = __builtin_amdgcn_wmma_f32_16x16x4_f32(
                    false, a[kk], false, bfr, (short)0, cf, false, false);
            cp = __builtin_amdgcn_wmma_f32_16x16x4_f32(
                    false, a[kk], false, bpr, (short)0, cp, false, false);
        }

        const float badd = bias[colBase + nn] + b_proj[colBase + nn];

#pragma unroll
        for (int v = 0; v < 8; ++v) {
            const int row = row0 + v + rowsel;
            if (row < n) {
                h  [(size_t)row * DIM + colBase + nn] = cf[v];
                out[(size_t)row * DIM + colBase + nn] = cp[v] + badd;
            }
        }
    }
}

// ---------------------------------------------------------------------------
// Kernel 2: edge scatter  out[dst] += val * h[src]
// One wave per edge; lane handles 4 contiguous floats (global_load_b128 +
// 4x global_atomic_add_f32, all L2-resident since h is 51MB < 192MB L2).
// ---------------------------------------------------------------------------
__global__ __launch_bounds__(256) void gcn_edge_scatter(
    const float* __restrict__ h,
    const int*   __restrict__ edge_src,
    const int*   __restrict__ edge_dst,
    const float* __restrict__ adj_vals,
    float* __restrict__ out,
    int e_total)
{
    const int e = blockIdx.x * (blockDim.x >> 5) + (threadIdx.x >> 5);
    if (e >= e_total) return;
    const int lane = threadIdx.x & 31;

    const int   src = edge_src[e];
    const int   dst = edge_dst[e];
    const float val = adj_vals[e];

    const float4 hv = ((const float4*)(h + (size_t)src * DIM))[lane];
    float* o = out + (size_t)dst * DIM + lane * 4;
    atomicAdd(o + 0, val * hv.x);
    atomicAdd(o + 1, val * hv.y);
    atomicAdd(o + 2, val * hv.z);
    atomicAdd(o + 3, val * hv.w);
}

// ---------------------------------------------------------------------------
// Kernel 3: out = tanh(out), float4 vectorized
// ---------------------------------------------------------------------------
__global__ __launch_bounds__(256) void gcn_tanh(float* __restrict__ out, int n4)
{
    const int i = blockIdx.x * blockDim.x + threadIdx.x;
    if (i >= n4) return;
    float4 v = ((float4*)out)[i];
    v.x = tanhf(v.x); v.y = tanhf(v.y); v.z = tanhf(v.z); v.w = tanhf(v.w);
    ((float4*)out)[i] = v;
}

extern "C" void kernel_launch(void* const* d_in, const int* in_sizes, int n_in,
                              void* d_out, int out_size, void* d_ws, size_t ws_size,
                              hipStream_t stream)
{
    // setup_inputs order: seq, edge_src, edge_dst, adj_vals, w_fc, w_proj, b_proj, bias
    const float* seq      = (const float*)d_in[0];
    const int*   edge_src = (const int*)  d_in[1];
    const int*   edge_dst = (const int*)  d_in[2];
    const float* adj_vals = (const float*)d_in[3];
    const float* w_fc     = (const float*)d_in[4];
    const float* w_proj   = (const float*)d_in[5];
    const float* b_proj   = (const float*)d_in[6];
    const float* bias     = (const float*)d_in[7];

    const int n = in_sizes[0] / DIM;   // 100000
    const int e = in_sizes[1];         // 1600000

    float* h   = (float*)d_ws;         // N*128 f32 scratch (51.2 MB)
    float* out = (float*)d_out;

    // 1) dual GEMM: writes h (scratch) and pre-activation residual into out
    {
        dim3 grid((n + 127) / 128);
        gcn_gemm_wmma<<<grid, 256, 0, stream>>>(seq, w_fc, w_proj, bias, b_proj,
                                                h, out, n);
    }
    // 2) scatter-add messages into out (wave per edge)
    {
        dim3 grid((e + 7) / 8);
        gcn_edge_scatter<<<grid, 256, 0, stream>>>(h, edge_src, edge_dst,
                                                   adj_vals, out, e);
    }
    // 3) tanh activation in place
    {
        const int n4 = n * (DIM / 4);
        dim3 grid((n4 + 255) / 256);
        gcn_tanh<<<grid, 256, 0, stream>>>(out, n4);
    }
}